// GAT_62234076119635
// MI455X (gfx1250) — compile-verified
//
#include <hip/hip_runtime.h>
#include <hip/hip_bf16.h>
#include <math.h>

typedef __attribute__((ext_vector_type(2))) float v2f;
typedef __attribute__((ext_vector_type(8))) float v8f;

#define NEG_SLOPE 0.2f

// Monotonic float<->uint mapping so we can use hardware u32 atomicMax for the
// segment-max (all encoded finite values are > 0, so init with 0 works since
// every node has a self-loop edge that always updates it).
__device__ __forceinline__ unsigned enc_f32(float f) {
  unsigned u = __float_as_uint(f);
  return u ^ ((u >> 31) ? 0xFFFFFFFFu : 0x80000000u);
}
__device__ __forceinline__ float dec_f32(unsigned u) {
  unsigned v = u ^ ((u >> 31) ? 0x80000000u : 0xFFFFFFFFu);
  return __uint_as_float(v);
}

// H[N,128] = X[N,128] @ W[128,128] using fp32 WMMA (16x16x4).
// Block = 256 threads = 8 waves; wave w owns the 16x16 tile at cols w*16;
// block owns 16 rows. K-loop of 32 steps of 4.
__global__ __launch_bounds__(256) void gat_gemm128(const float* __restrict__ X,
                                                   const float* __restrict__ W,
                                                   float* __restrict__ Hout,
                                                   int N) {
  const int lane  = threadIdx.x & 31;
  const int wv    = threadIdx.x >> 5;      // 0..7 -> column tile
  const int m0    = blockIdx.x << 4;       // 16 rows per block
  const int n0    = wv << 4;
  const int r     = lane & 15;
  const int khalf = (lane >> 4) << 1;      // 0 for lanes 0-15, 2 for lanes 16-31

  int arow = m0 + r;
  if (arow >= N) arow = N - 1;             // duplicate-load guard (stores guarded below)
  const float* ap = X + (size_t)arow * 128;

  v8f acc = {0.f, 0.f, 0.f, 0.f, 0.f, 0.f, 0.f, 0.f};
#pragma unroll
  for (int k0 = 0; k0 < 128; k0 += 4) {
    const int ka = k0 + khalf;
    // A 16x4 tile (ISA layout): lane half selects K pair, VGPR index selects K lsb
    v2f a; a.x = ap[ka]; a.y = ap[ka + 1];
    // B 4x16 tile: VGPR v holds row K = k0 + khalf + v, col = n0 + (lane&15)
    v2f b; b.x = W[(size_t)ka * 128 + n0 + r];
           b.y = W[(size_t)(ka + 1) * 128 + n0 + r];
    acc = __builtin_amdgcn_wmma_f32_16x16x4_f32(false, a, false, b,
                                                (short)0, acc, false, false);
  }
  // C/D layout: VGPR i -> row (m0 + i + 8*lanehalf), col n0 + (lane&15)
  const int rowbase = m0 + ((lane >> 4) << 3);
  const int col     = n0 + r;
#pragma unroll
  for (int i = 0; i < 8; ++i) {
    const int row = rowbase + i;
    if (row < N) Hout[(size_t)row * 128 + col] = acc[i];
  }
}

// Per (node, head): es/ed attention dots; also init m(-inf enc), s=0, acc=0.
__global__ void gat_scores_init(const float* __restrict__ Hm,
                                const float* __restrict__ asrc,
                                const float* __restrict__ adst,
                                float* __restrict__ es, float* __restrict__ ed,
                                unsigned* __restrict__ menc,
                                float* __restrict__ ssum,
                                float* __restrict__ acc, int N) {
  const int idx = blockIdx.x * blockDim.x + threadIdx.x;
  if (idx >= N * 4) return;
  const int hd = idx & 3;
  const size_t base = (size_t)(idx >> 2) * 128 + hd * 32;
  const float* hp  = Hm + base;
  float*       ap  = acc + base;
  const float* sa  = asrc + hd * 32;
  const float* da  = adst + hd * 32;
  float s = 0.f, d = 0.f;
#pragma unroll
  for (int c = 0; c < 32; c += 4) {
    float4 hv = *(const float4*)(hp + c);
    float4 av = *(const float4*)(sa + c);
    float4 bv = *(const float4*)(da + c);
    s += hv.x * av.x + hv.y * av.y + hv.z * av.z + hv.w * av.w;
    d += hv.x * bv.x + hv.y * bv.y + hv.z * bv.z + hv.w * bv.w;
    *(float4*)(ap + c) = make_float4(0.f, 0.f, 0.f, 0.f);
  }
  es[idx] = s;
  ed[idx] = d;
  menc[idx] = 0u;
  ssum[idx] = 0.f;
}

__device__ __forceinline__ float lrelu(float x) {
  return x > 0.f ? x : NEG_SLOPE * x;
}

// Pass 1: per-edge segment max of leaky_relu(es[src]+ed[dst]) into menc[dst].
__global__ void gat_edge_max(const int* __restrict__ ei,
                             const float* __restrict__ es,
                             const float* __restrict__ ed,
                             unsigned* __restrict__ menc, int E, int N) {
  const int e = blockIdx.x * blockDim.x + threadIdx.x;
  const int Et = E + N;
  if (e >= Et) return;
  int s, d;
  if (e < E) { s = ei[e]; d = ei[E + e]; } else { s = d = e - E; }
  float4 vs = *(const float4*)(es + (size_t)s * 4);
  float4 vd = *(const float4*)(ed + (size_t)d * 4);
  unsigned* mp = menc + (size_t)d * 4;
  atomicMax(mp + 0, enc_f32(lrelu(vs.x + vd.x)));
  atomicMax(mp + 1, enc_f32(lrelu(vs.y + vd.y)));
  atomicMax(mp + 2, enc_f32(lrelu(vs.z + vd.z)));
  atomicMax(mp + 3, enc_f32(lrelu(vs.w + vd.w)));
}

// Pass 2: one wave per edge. ex = exp(l - m[dst]); s[dst]+=ex;
// acc[dst, :] += ex * h[src, :]  (lane covers 4 channels; head = lane>>3).
__global__ __launch_bounds__(256) void gat_edge_scatter(
    const int* __restrict__ ei,
    const float* __restrict__ es, const float* __restrict__ ed,
    const unsigned* __restrict__ menc, float* __restrict__ ssum,
    const float* __restrict__ Hm, float* __restrict__ acc, int E, int N) {
  const int e = blockIdx.x * 8 + (threadIdx.x >> 5);
  const int lane = threadIdx.x & 31;
  const int Et = E + N;
  if (e >= Et) return;
  int s, d;
  if (e < E) { s = ei[e]; d = ei[E + e]; } else { s = d = e - E; }
  const int hd = lane >> 3;
  float l = es[(size_t)s * 4 + hd] + ed[(size_t)d * 4 + hd];
  l = lrelu(l);
  const float mh = dec_f32(menc[(size_t)d * 4 + hd]);
  const float ex = expf(l - mh);
  if ((lane & 7) == 0) atomicAdd(&ssum[(size_t)d * 4 + hd], ex);
  float4 hv = *(const float4*)(Hm + (size_t)s * 128 + lane * 4);
  float* op = acc + (size_t)d * 128 + lane * 4;
  atomicAdd(op + 0, hv.x * ex);
  atomicAdd(op + 1, hv.y * ex);
  atomicAdd(op + 2, hv.z * ex);
  atomicAdd(op + 3, hv.w * ex);
}

// Pass 3: normalize by s[dst], add bias, optional ELU.
__global__ void gat_finalize(const float* __restrict__ acc,
                             const float* __restrict__ ssum,
                             const float* __restrict__ bias,
                             float* __restrict__ outp, int N, int apply_elu) {
  const int idx = blockIdx.x * blockDim.x + threadIdx.x;
  if (idx >= N * 128) return;
  const int n  = idx >> 7;
  const int hd = (idx >> 5) & 3;
  float v = acc[idx] / ssum[(size_t)n * 4 + hd] + bias[idx & 127];
  if (apply_elu) v = v > 0.f ? v : expm1f(v);
  outp[idx] = v;
}

extern "C" void kernel_launch(void* const* d_in, const int* in_sizes, int n_in,
                              void* d_out, int out_size, void* d_ws, size_t ws_size,
                              hipStream_t stream) {
  const float* x  = (const float*)d_in[0];
  const int*   ei = (const int*)d_in[1];
  const int N = in_sizes[0] / 128;
  const int E = in_sizes[1] / 2;

  const float* Wl[3] = {(const float*)d_in[2], (const float*)d_in[6],  (const float*)d_in[10]};
  const float* As[3] = {(const float*)d_in[3], (const float*)d_in[7],  (const float*)d_in[11]};
  const float* Ad[3] = {(const float*)d_in[4], (const float*)d_in[8],  (const float*)d_in[12]};
  const float* Bb[3] = {(const float*)d_in[5], (const float*)d_in[9],  (const float*)d_in[13]};

  float*    F0   = (float*)d_ws;
  float*    F1   = F0 + (size_t)N * 128;
  float*    es   = F1 + (size_t)N * 128;
  float*    ed   = es + (size_t)N * 4;
  unsigned* menc = (unsigned*)(ed + (size_t)N * 4);
  float*    ssum = (float*)(menc + (size_t)N * 4);

  const int Et = E + N;
  const dim3 bGemm(256);
  const dim3 gGemm((N + 15) / 16);
  const int gScore = (N * 4 + 255) / 256;
  const int gMax   = (Et + 255) / 256;
  const int gScat  = (Et + 7) / 8;
  const int gFin   = (N * 128 + 255) / 256;

  const float* inp = x;
  for (int layer = 0; layer < 3; ++layer) {
    float* Hm = (layer & 1) ? F1 : F0;
    float* Ac = (layer & 1) ? F0 : F1;
    gat_gemm128<<<gGemm, bGemm, 0, stream>>>(inp, Wl[layer], Hm, N);
    gat_scores_init<<<gScore, 256, 0, stream>>>(Hm, As[layer], Ad[layer],
                                                es, ed, menc, ssum, Ac, N);
    gat_edge_max<<<gMax, 256, 0, stream>>>(ei, es, ed, menc, E, N);
    gat_edge_scatter<<<gScat, 256, 0, stream>>>(ei, es, ed, menc, ssum, Hm, Ac, E, N);
    float* ot = (layer == 2) ? (float*)d_out : Hm;
    gat_finalize<<<gFin, 256, 0, stream>>>(Ac, ssum, Bb[layer], ot, N,
                                           layer < 2 ? 1 : 0);
    inp = Hm;
  }
}